// InputBlock_26036091748438
// MI455X (gfx1250) — compile-verified
//
#include <hip/hip_runtime.h>

// Shapes (hardcoded from reference): B=2, D=16, H=48, W=48
//   Q = 16*48*48 = 36864, HW = 2304, voxels per batch = 2304
// Phase 1 (attn_stack_kernel): 4-layer gated channel mix via
//   V_WMMA_F32_16X16X4_F32; emits candidate table float4{x|BIG, y, z, val}
//   directly into d_ws (mask folded into x coordinate: +1e10 => d ~= BIG).
// Phase 2 (idw_kernel): exact top-4 nearest neighbor, double-buffered
//   GLOBAL_LOAD_ASYNC_TO_LDS_B128 staging (ASYNCcnt), branchless 4-deep
//   compare-exchange on squared distance.

typedef __attribute__((ext_vector_type(2))) float v2f;
typedef __attribute__((ext_vector_type(8))) float v8f;
typedef __attribute__((__vector_size__(16))) int v4i;
typedef __attribute__((address_space(1))) v4i* gptr_v4i;   // global int4*
typedef __attribute__((address_space(3))) v4i* lptr_v4i;   // LDS int4*

#define HW    2304
#define QTOT  36864
#define NLAY  4
#define CH    1024          // candidates per chunk (16 KB); QTOT/CH = 36 chunks
#define NCHUNK (QTOT / CH)

#if __has_builtin(__builtin_amdgcn_global_load_async_to_lds_b128) && \
    __has_builtin(__builtin_amdgcn_s_wait_asynccnt)
#define HAVE_ASYNC_LDS 1
#else
#define HAVE_ASYNC_LDS 0
#endif

// ---------------------------------------------------------------------------
// Kernel A: channel-mixing stack + candidate-table emit.
// One wave (32 lanes) owns a tile of 16 voxels x 16 channels; 288 tiles ->
// 36 blocks x 8 waves. No divergence (EXEC all-ones, required by WMMA).
// ---------------------------------------------------------------------------
__global__ __launch_bounds__(256) void attn_stack_kernel(
    const float* __restrict__ in,    // [2][16][48][48]
    const float* __restrict__ cw,    // [4][16][16]  (l, o, i)
    const float* __restrict__ cb,    // [4][16]
    const int*   __restrict__ mask,  // [2][16][48][48]
    float4*      __restrict__ cand)  // [2][36864] {x|BIG, y, z, val}
{
  __shared__ float xs[8][16][17];           // per-wave relayout tile (C -> A)

  const int lane = threadIdx.x & 31;
  const int wv   = threadIdx.x >> 5;
  const int tile = blockIdx.x * 8 + wv;     // 0..287
  const int b    = tile / 144;
  const int v0   = (tile % 144) * 16;       // first voxel of tile
  const int half = lane >> 4;               // 0|1
  const int nn   = lane & 15;               // N column / channel (== z index)

  // B fragments: B[i][o] = cw[l][o][i]; B 4x16 layout:
  //   vgpr j, lanes 0-15: K=j ; lanes 16-31: K=j+2 ; N = lane&15
  v2f bf[NLAY][4];
#pragma unroll
  for (int l = 0; l < NLAY; ++l)
#pragma unroll
    for (int kk = 0; kk < 4; ++kk)
#pragma unroll
      for (int j = 0; j < 2; ++j)
        bf[l][kk][j] = cw[l * 256 + nn * 16 + kk * 4 + j + half * 2];

  float bias[NLAY];
#pragma unroll
  for (int l = 0; l < NLAY; ++l) bias[l] = cb[l * 16 + nn];

  // x in C/D layout: x_c[j] = element (m = j + 8*half, n = nn)
  float x_c[8];
  const float* inb = in + b * QTOT + v0;
#pragma unroll
  for (int j = 0; j < 8; ++j)
    x_c[j] = inb[nn * HW + j + 8 * half];

#pragma unroll
  for (int l = 0; l < NLAY; ++l) {
    // C-layout -> LDS tile xs[m][i]
#pragma unroll
    for (int j = 0; j < 8; ++j)
      xs[wv][j + 8 * half][nn] = x_c[j];
    __syncthreads();

    // gate = x @ W^T via 4 chained V_WMMA_F32_16X16X4_F32 (K = 16)
    v8f acc = {};
#pragma unroll
    for (int kk = 0; kk < 4; ++kk) {
      v2f af;   // A 16x4: M = nn, K = j + 2*half within slice kk
#pragma unroll
      for (int j = 0; j < 2; ++j)
        af[j] = xs[wv][nn][kk * 4 + j + half * 2];
      acc = __builtin_amdgcn_wmma_f32_16x16x4_f32(
          /*neg_a=*/false, af, /*neg_b=*/false, bf[l][kk],
          /*c_mod=*/(short)0, acc, /*reuse_a=*/false, /*reuse_b=*/false);
    }
    __syncthreads();   // reads done before next layer's LDS overwrite

    // x = relu(x + x*gate)
#pragma unroll
    for (int j = 0; j < 8; ++j) {
      float g = acc[j] + bias[l];
      x_c[j] = fmaxf(fmaf(x_c[j], g, x_c[j]), 0.0f);
    }
  }

  // Emit candidate table: q = d*HW + vox ; d = nn (also lattice z index)
  const float s47 = 1.0f / 47.0f, s15 = 1.0f / 15.0f;
  const float gz  = nn * s15;
#pragma unroll
  for (int j = 0; j < 8; ++j) {
    const int vox = v0 + j + 8 * half;
    const int cy  = vox / 48;
    const int cx  = vox - cy * 48;
    const int q   = b * QTOT + nn * HW + vox;
    const float big = mask[q] ? 0.0f : 1e10f;   // masked-out -> d ~= BIG
    cand[q] = make_float4(cx * s47 + big, cy * s47, gz, x_c[j]);
  }
}

// ---------------------------------------------------------------------------
// Kernel B helpers
// ---------------------------------------------------------------------------
__device__ __forceinline__ void stage_chunk(const float4* __restrict__ src,
                                            float4* __restrict__ dst_lds,
                                            int tid) {
#if HAVE_ASYNC_LDS
#pragma unroll
  for (int k = 0; k < CH; k += 256) {
    __builtin_amdgcn_global_load_async_to_lds_b128(
        (gptr_v4i)(src + k + tid),      // global src (addrspace 1, v4i*)
        (lptr_v4i)(dst_lds + k + tid),  // LDS dst   (addrspace 3, v4i*)
        /*offset=*/0, /*cpol=*/0);
  }
#else
#pragma unroll
  for (int k = 0; k < CH; k += 256) {
    dst_lds[k + tid] = src[k + tid];
    __builtin_prefetch(src + k + tid + CH, 0, 3);   // global_prefetch_b8
  }
#endif
}

__device__ __forceinline__ void consume_chunk(
    const float4* __restrict__ cptr, float fx, float fy, float fz,
    float& bd0, float& bd1, float& bd2, float& bd3,
    float& bv0, float& bv1, float& bv2, float& bv3) {
#pragma unroll 4
  for (int j = 0; j < CH; ++j) {
    const float4 c = cptr[j];                        // ds_load_b128 broadcast
    const float dx = fx - c.x, dy = fy - c.y, dz = fz - c.z;
    const float d2 = fmaf(dx, dx, fmaf(dy, dy, dz * dz));
    const float nv = c.w;
    // evict current max if beaten, then bubble up (branchless)
    {
      const bool cc = d2 < bd3;
      bd3 = cc ? d2 : bd3;  bv3 = cc ? nv : bv3;
    }
    {
      const bool cc = bd3 < bd2;  const float td = bd2, tv = bv2;
      bd2 = cc ? bd3 : bd2;  bv2 = cc ? bv3 : bv2;
      bd3 = cc ? td  : bd3;  bv3 = cc ? tv  : bv3;
    }
    {
      const bool cc = bd2 < bd1;  const float td = bd1, tv = bv1;
      bd1 = cc ? bd2 : bd1;  bv1 = cc ? bv2 : bv1;
      bd2 = cc ? td  : bd2;  bv2 = cc ? tv  : bv2;
    }
    {
      const bool cc = bd1 < bd0;  const float td = bd0, tv = bv0;
      bd0 = cc ? bd1 : bd0;  bv0 = cc ? bv1 : bv0;
      bd1 = cc ? td  : bd1;  bv1 = cc ? tv  : bv1;
    }
  }
}

// ---------------------------------------------------------------------------
// Kernel B: IDW with exact top-4 over all candidates.
// One query per lane; candidates double-buffered into LDS (2 x 16 KB) via
// async global->LDS loads so global traffic overlaps the select-heavy VALU
// inner loop. Peeled epilogue drains ASYNCcnt to 0.
// ---------------------------------------------------------------------------
__global__ __launch_bounds__(256) void idw_kernel(
    const float4* __restrict__ cand,  // [2][36864]
    float*        __restrict__ out)   // [2][36864]
{
  __shared__ float4 buf[2][CH];              // 32 KB double buffer

  const int tid = threadIdx.x;
  const int b   = blockIdx.x / 144;
  const int q   = (blockIdx.x % 144) * 256 + tid;

  const float s47 = 1.0f / 47.0f, s15 = 1.0f / 15.0f;
  int qz = q / HW;  int r = q - qz * HW;
  int qy = r / 48;  int qx = r - qy * 48;
  const float fx = qx * s47, fy = qy * s47, fz = qz * s15;

  float bd0 = 1e30f, bd1 = 1e30f, bd2 = 1e30f, bd3 = 1e30f;
  float bv0 = 0.f,   bv1 = 0.f,   bv2 = 0.f,   bv3 = 0.f;

  const float4* cb4 = cand + b * QTOT;

  // prologue: stage chunk 0
  stage_chunk(cb4, &buf[0][0], tid);

  for (int ci = 0; ci < NCHUNK - 1; ++ci) {
    const int bi = ci & 1;
    stage_chunk(cb4 + (ci + 1) * CH, &buf[bi ^ 1][0], tid);  // next chunk
#if HAVE_ASYNC_LDS
    __builtin_amdgcn_s_wait_asynccnt(4);   // chunk ci's 4 loads done
#endif
    __syncthreads();
    consume_chunk(&buf[bi][0], fx, fy, fz,
                  bd0, bd1, bd2, bd3, bv0, bv1, bv2, bv3);
    __syncthreads();
  }
  // peeled last chunk: drain ASYNCcnt fully
#if HAVE_ASYNC_LDS
  __builtin_amdgcn_s_wait_asynccnt(0);
#endif
  __syncthreads();
  consume_chunk(&buf[(NCHUNK - 1) & 1][0], fx, fy, fz,
                bd0, bd1, bd2, bd3, bv0, bv1, bv2, bv3);

  // IDW weights (rho == 2 fast path), matching reference epilogue
  const float d0 = sqrtf(bd0), d1 = sqrtf(bd1), d2 = sqrtf(bd2), d3 = sqrtf(bd3);
  const float i0 = 1.0f / (d0 + 0.05f), i1 = 1.0f / (d1 + 0.05f);
  const float i2 = 1.0f / (d2 + 0.05f), i3 = 1.0f / (d3 + 0.05f);
  const float w0 = i0 * i0, w1 = i1 * i1, w2 = i2 * i2, w3 = i3 * i3;
  const float s  = w0 + w1 + w2 + w3 + 1e-12f;
  out[b * QTOT + q] =
      (bv0 * w0 + bv1 * w1 + bv2 * w2 + bv3 * w3) / s;
}

// ---------------------------------------------------------------------------
extern "C" void kernel_launch(void* const* d_in, const int* in_sizes, int n_in,
                              void* d_out, int out_size, void* d_ws, size_t ws_size,
                              hipStream_t stream) {
  const float* inp    = (const float*)d_in[0];  // [2,16,48,48] f32
  const int*   mask   = (const int*)  d_in[1];  // [2,16,48,48] i32
  const float* conv_w = (const float*)d_in[2];  // [4,16,16]    f32
  const float* conv_b = (const float*)d_in[3];  // [4,16]       f32
  float*       out    = (float*)d_out;          // [2,16,48,48] f32
  float4*      cand   = (float4*)d_ws;          // [2][36864] candidate table

  // Phase 1: 288 wave-tiles -> 36 blocks x 256 threads (8 waves each)
  attn_stack_kernel<<<36, 256, 0, stream>>>(inp, conv_w, conv_b, mask, cand);

  // Phase 2: one query per thread -> 2 * 36864 / 256 = 288 blocks
  idw_kernel<<<288, 256, 0, stream>>>(cand, out);
}